// GenAttentionMask_32384053412245
// MI455X (gfx1250) — compile-verified
//
#include <hip/hip_runtime.h>

// 128-bit vector type usable with clang's nontemporal builtins.
typedef float v4f __attribute__((ext_vector_type(4)));

#define MAXSEQ          2048
#define TPB             256
#define ROWS_PER_BLOCK  16

__global__ __launch_bounds__(TPB) void GenAttentionMask_kernel(
    const float* __restrict__ mask,        // [BATCH, 1, 2048, 2048] fp32
    const int*   __restrict__ seq_lengths, // [BATCH]
    const int*   __restrict__ head_num_p,  // scalar
    float*       __restrict__ out)         // packed bool-as-float
{
    const int b = blockIdx.z;

    // Uniform scalar work: this batch's seq length + packed output base offset.
    const int hn = *head_num_p;
    long long base = 0;
    for (int bb = 0; bb < b; ++bb) {
        long long sp = (long long)seq_lengths[bb];
        base += sp * sp;
    }
    base *= (long long)hn;

    const int s  = seq_lengths[b];
    const int q4 = s >> 2;                       // float4 quads per row
    const int j4 = blockIdx.x * TPB + threadIdx.x;
    if (j4 >= q4) return;                        // column quad outside this batch's block
    const int i0 = blockIdx.y * ROWS_PER_BLOCK;
    if (i0 >= s) return;                         // row strip outside this batch's block

    const float* src = mask + (size_t)b * MAXSEQ * MAXSEQ
                            + (size_t)i0 * MAXSEQ
                            + (size_t)j4 * 4;
    const long long ss = (long long)s * (long long)s;

    int rmax = s - i0;
    if (rmax > ROWS_PER_BLOCK) rmax = ROWS_PER_BLOCK;

    const _Float16 half_pt5 = (_Float16)0.5f;

    for (int r = 0; r < rmax; ++r) {
        // gfx1250 prefetch path (global_prefetch_b8) for the next row's quad.
        if (r + 1 < rmax)
            __builtin_prefetch((const void*)(src + (size_t)(r + 1) * MAXSEQ), 0, 0);

        // Read-once: NT 128-bit load (th:TH_LOAD_NT), don't pollute L2.
        v4f x = __builtin_nontemporal_load((const v4f*)(src + (size_t)r * MAXSEQ));

        // Match reference exactly: fp32 -> fp16 (RNE) then compare > 0.5 in fp16.
        v4f o;
        o.x = ((_Float16)x.x > half_pt5) ? 1.0f : 0.0f;
        o.y = ((_Float16)x.y > half_pt5) ? 1.0f : 0.0f;
        o.z = ((_Float16)x.z > half_pt5) ? 1.0f : 0.0f;
        o.w = ((_Float16)x.w > half_pt5) ? 1.0f : 0.0f;

        const long long row_off = base + (long long)(i0 + r) * (long long)s
                                       + (long long)j4 * 4;

        // Write 8x from registers: NT 128-bit stores (th:TH_STORE_NT),
        // streaming output must not thrash the 192 MB L2.
        float* op = out + row_off;
        for (int h = 0; h < hn; ++h) {
            __builtin_nontemporal_store(o, (v4f*)op);
            op += ss;
        }
    }
}

extern "C" void kernel_launch(void* const* d_in, const int* in_sizes, int n_in,
                              void* d_out, int out_size, void* d_ws, size_t ws_size,
                              hipStream_t stream) {
    const float* mask  = (const float*)d_in[0];
    const int*   slens = (const int*)d_in[1];
    const int*   hnum  = (const int*)d_in[2];
    float*       out   = (float*)d_out;

    const int batch = in_sizes[1];   // number of seq_lengths entries

    dim3 block(TPB, 1, 1);
    dim3 grid((MAXSEQ / 4 + TPB - 1) / TPB,   // 2 blocks of column-quads
              MAXSEQ / ROWS_PER_BLOCK,        // 128 row strips
              batch);                         // 8 batches
    GenAttentionMask_kernel<<<grid, block, 0, stream>>>(mask, slens, hnum, out);
}